// Lumina2RotaryPosEmbed_70841190580229
// MI455X (gfx1250) — compile-verified
//
#include <hip/hip_runtime.h>
#include <stdint.h>

// ---------------------------------------------------------------------------
// Lumina2 rotary-pos-embed preprocessing for MI455X (gfx1250).
// Memory-bound (~180 MB traffic => ~8us @ 23.3 TB/s). No matmul in the
// workload, so no WMMA; the CDNA5-specific path used is the async
// global->LDS DMA (global_load_async_to_lds_b128 + s_wait_asynccnt) to stage
// the patchify transpose with wide coalesced reads AND writes.
// ---------------------------------------------------------------------------

namespace {
constexpr int Bn = 32, Cn = 16, Hn = 128, Wn = 128;
constexpr int Pp = 2, HT = Hn / Pp, WT = Wn / Pp;      // 64, 64
constexpr int IMG_LEN = HT * WT;                       // 4096
constexpr int LC = 256;                                // caption buffer len
constexpr int ND = 48;                                 // sum(axes_dim)/2
constexpr int SMAX = LC + IMG_LEN;                     // 4352 >= max_seq_len
// -log2(10000)/16  (inv_freq_j = 2^(j * this))
constexpr float NEG_L2T_16 = -0.83048202372184056309f;

// flat float32 offsets into d_out (reference tuple order)
constexpr size_t O0 = 0;                               // padded_img_embed
constexpr size_t N0 = (size_t)Bn * IMG_LEN * 64;       // 8,388,608
constexpr size_t O1 = O0 + N0;                         // padded_img_mask
constexpr size_t N1 = (size_t)Bn * IMG_LEN;            // 131,072
constexpr size_t O2 = O1 + N1;                         // img_sizes [32,2]
constexpr size_t O3 = O2 + 2 * Bn;                     // cap_lens  [32]
constexpr size_t O4 = O3 + Bn;                         // l_effective [32]
constexpr size_t O5 = O4 + Bn;                         // freqs_cis (B*S*96)
constexpr size_t N6 = (size_t)Bn * LC * ND * 2;        // cap_freqs floats
constexpr size_t N7 = (size_t)Bn * IMG_LEN * ND * 2;   // img_freqs floats

__device__ __forceinline__ float inv_freq(int j) {
  return exp2f((float)j * NEG_L2T_16);                 // 10000^(-j/16)
}
} // namespace

// ---------------------------------------------------------------------------
// Kernel A: cap_lens, max_cap -> ws; small scalar outputs.
// ws layout (int32): [0..31] cap_lens, [32] max_cap
// ---------------------------------------------------------------------------
__global__ __launch_bounds__(32) void k_caps(const int* __restrict__ mask,
                                             float* __restrict__ out,
                                             int* __restrict__ ws) {
  __shared__ int sc[Bn];
  const int b = threadIdx.x;                 // 0..31
  const int* row = mask + (size_t)b * LC;
  int sum = 0;
  #pragma unroll 8
  for (int i = 0; i < LC; ++i) sum += (row[i] != 0) ? 1 : 0;
  sc[b] = sum;
  ws[b] = sum;
  out[O3 + b] = (float)sum;                  // cap_lens
  out[O4 + b] = (float)IMG_LEN;              // l_effective_img_len
  out[O2 + 2 * b + 0] = (float)Hn;           // img_sizes
  out[O2 + 2 * b + 1] = (float)Wn;
  __syncthreads();
  if (b == 0) {
    int m = 0;
    for (int i = 0; i < Bn; ++i) m = max(m, sc[i]);
    ws[32] = m;
    const size_t S = (size_t)m + IMG_LEN;    // max_seq_len
    const size_t o8 = O5 + (size_t)Bn * S * (ND * 2) + N6 + N7;
    out[o8] = (float)S;                      // scalar max_seq_len
  }
}

// ---------------------------------------------------------------------------
// Kernel B: patchify [B,C,H,W] -> [B, 4096, 64] via async global->LDS stage,
// plus padded_img_mask = 1.0. One block per (b, ht): 16KB in, 16KB out.
// ---------------------------------------------------------------------------
#define CHUNK_F 132   // 128 floats of data + 4-float pad (bank-conflict dodge)

__global__ __launch_bounds__(256) void k_patchify(const float* __restrict__ in,
                                                  float* __restrict__ out) {
  __shared__ __align__(16) float lds[32 * CHUNK_F];    // 16,896 B

  const int t   = threadIdx.x;
  const int blk = blockIdx.x;                          // b*HT + ht
  const int b   = blk >> 6;
  const int ht  = blk & 63;

  // ---- async DMA: 32 chunks (c in 0..15, i in 0..1) of 512B each ----
  const int m     = t >> 3;                            // chunk 0..31
  const int c     = m >> 1;
  const int i     = m & 1;
  const int lane8 = t & 7;
  const float* gsrc =
      in + (((size_t)(b * Cn + c) * Hn + (2 * ht + i)) * Wn);  // row of 128 f
  const uint32_t lds_chunk =
      (uint32_t)(uintptr_t)(&lds[0]) + (uint32_t)(m * CHUNK_F * 4);
  #pragma unroll
  for (int q = 0; q < 4; ++q) {
    const int seg = q * 8 + lane8;                     // 16B segment 0..31
    uint64_t ga = (uint64_t)(uintptr_t)gsrc + (uint64_t)(seg * 16);
    uint32_t la = lds_chunk + (uint32_t)(seg * 16);
    asm volatile("global_load_async_to_lds_b128 %0, %1, off"
                 :: "v"(la), "v"(ga) : "memory");
  }
  asm volatile("s_wait_asynccnt 0" ::: "memory");
  __syncthreads();

  // ---- emit: thread t -> (wt, pi, pj), 16 contiguous floats (c fastest) ----
  const int wt = t >> 2;
  const int pi = (t >> 1) & 1;
  const int pj = t & 1;
  float v[16];
  #pragma unroll
  for (int cc = 0; cc < 16; ++cc)
    v[cc] = lds[(cc * 2 + pi) * CHUNK_F + 2 * wt + pj];

  float4* dst = (float4*)(out + O0 +
      ((size_t)(b * IMG_LEN + ht * WT + wt) * 64 + (pi * 2 + pj) * 16));
  dst[0] = make_float4(v[0],  v[1],  v[2],  v[3]);
  dst[1] = make_float4(v[4],  v[5],  v[6],  v[7]);
  dst[2] = make_float4(v[8],  v[9],  v[10], v[11]);
  dst[3] = make_float4(v[12], v[13], v[14], v[15]);

  if (t < WT)                                          // mask: 64 tokens/row
    out[O1 + (size_t)b * IMG_LEN + ht * WT + t] = 1.0f;
}

// ---------------------------------------------------------------------------
// Kernel C: freqs_cis [B, S, 48] complex; pad region (pos=0) -> (1,0).
// ---------------------------------------------------------------------------
__global__ __launch_bounds__(256) void k_freqs(const int* __restrict__ ws,
                                               float* __restrict__ out) {
  const size_t gid = (size_t)blockIdx.x * blockDim.x + threadIdx.x;
  const int k = (int)(gid % ND);
  const size_t r = gid / ND;
  const int s = (int)(r % SMAX);
  const int b = (int)(r / SMAX);
  const int S = ws[32] + IMG_LEN;
  if (s >= S) return;
  const int cap = ws[b];

  const int axis = k >> 4, j = k & 15;
  int pos = 0;
  if (s < cap) {
    pos = (axis == 0) ? s : 0;
  } else if (s < cap + IMG_LEN) {
    const int rel = s - cap;
    pos = (axis == 0) ? cap : ((axis == 1) ? (rel >> 6) : (rel & 63));
  }
  const float ang = (float)pos * inv_freq(j);
  float sn, cs;
  __sincosf(ang, &sn, &cs);
  float2* dst = (float2*)(out + O5);
  dst[((size_t)b * S + s) * ND + k] = make_float2(cs, sn);
}

// ---------------------------------------------------------------------------
// Kernel D: cap_freqs_cis [B, 256, 48]; t<cap: axis0=cis(t*f), axes1/2=(1,0);
// else complex zero.
// ---------------------------------------------------------------------------
__global__ __launch_bounds__(256) void k_capfreqs(const int* __restrict__ ws,
                                                  float* __restrict__ out) {
  const int gid = blockIdx.x * 256 + threadIdx.x;      // < 32*256*48
  const int k = gid % ND;
  const int r = gid / ND;
  const int t = r % LC;
  const int b = r / LC;
  const int S = ws[32] + IMG_LEN;
  const int cap = ws[b];

  float cs = 0.0f, sn = 0.0f;
  if (t < cap) {
    if ((k >> 4) == 0) {
      const float ang = (float)t * inv_freq(k & 15);
      __sincosf(ang, &sn, &cs);
    } else { cs = 1.0f; sn = 0.0f; }
  }
  const size_t o6 = O5 + (size_t)Bn * S * (ND * 2);
  float2* dst = (float2*)(out + o6);
  dst[((size_t)b * LC + t) * ND + k] = make_float2(cs, sn);
}

// ---------------------------------------------------------------------------
// Kernel E: img_freqs_cis [B, 4096, 48]: axis0=cis(cap*f), axis1=cis((i/64)f),
// axis2=cis((i%64)f).
// ---------------------------------------------------------------------------
__global__ __launch_bounds__(256) void k_imgfreqs(const int* __restrict__ ws,
                                                  float* __restrict__ out) {
  const size_t gid = (size_t)blockIdx.x * 256 + threadIdx.x;  // < 32*4096*48
  const int k = (int)(gid % ND);
  const size_t r = gid / ND;
  const int ii = (int)(r % IMG_LEN);
  const int b  = (int)(r / IMG_LEN);
  const int S = ws[32] + IMG_LEN;
  const int cap = ws[b];

  const int axis = k >> 4, j = k & 15;
  const int pos = (axis == 0) ? cap : ((axis == 1) ? (ii >> 6) : (ii & 63));
  const float ang = (float)pos * inv_freq(j);
  float sn, cs;
  __sincosf(ang, &sn, &cs);

  const size_t o7 = O5 + (size_t)Bn * S * (ND * 2) + N6;
  float2* dst = (float2*)(out + o7);
  dst[((size_t)b * IMG_LEN + ii) * ND + k] = make_float2(cs, sn);
}

// ---------------------------------------------------------------------------
extern "C" void kernel_launch(void* const* d_in, const int* in_sizes, int n_in,
                              void* d_out, int out_size, void* d_ws, size_t ws_size,
                              hipStream_t stream) {
  const float* hs  = (const float*)d_in[0];   // [32,16,128,128]
  const int*   msk = (const int*)d_in[1];     // [32,256] (bool as int32)
  float* out = (float*)d_out;
  int*   ws  = (int*)d_ws;

  k_caps<<<1, 32, 0, stream>>>(msk, out, ws);

  k_patchify<<<Bn * HT, 256, 0, stream>>>(hs, out);            // 2048 blocks

  const int nfreq = (Bn * SMAX * ND) / 256;                    // 26112
  k_freqs<<<nfreq, 256, 0, stream>>>(ws, out);

  const int ncap = (Bn * LC * ND) / 256;                       // 1536
  k_capfreqs<<<ncap, 256, 0, stream>>>(ws, out);

  const int nimg = (Bn * IMG_LEN * ND) / 256;                  // 24576
  k_imgfreqs<<<nimg, 256, 0, stream>>>(ws, out);
}